// MGEPC_15453292331335
// MI455X (gfx1250) — compile-verified
//
#include <hip/hip_runtime.h>
#include <hip/hip_bf16.h>

// MI455X / gfx1250, wave32. Dense layers use V_WMMA_F32_16X16X4_F32 (full f32,
// matching the f32 reference; workload is bandwidth-bound so lower precision
// buys nothing). Each wave computes 4 16x16 tiles sharing one B fragment.
// SpMM uses global_atomic_add_f32 scatter.

typedef float v2f __attribute__((ext_vector_type(2)));
typedef float v8f __attribute__((ext_vector_type(8)));

#define HDIM 128

// ---------------------------------------------------------------- utilities
__global__ void zero_f32_kernel(float* __restrict__ p, long long n) {
  long long i = (long long)blockIdx.x * blockDim.x + threadIdx.x;
  if (i < n) p[i] = 0.f;
}

__global__ void degree_accum_kernel(const int* __restrict__ dst, float* __restrict__ deg, int E) {
  int e = blockIdx.x * blockDim.x + threadIdx.x;
  if (e < E) atomicAdd(&deg[dst[e]], 1.0f);
}

__global__ void finish_dinv_kernel(float* __restrict__ deg, int n) {
  int i = blockIdx.x * blockDim.x + threadIdx.x;
  if (i < n) {
    float d = deg[i];
    d = d < 1.f ? 1.f : d;
    deg[i] = rsqrtf(d);
  }
}

// out[:, 0:128](ldc) = bias  (used when theta0 == 0: GEMM degenerates to bias)
__global__ void fill_bias_kernel(float* __restrict__ out, int ldc,
                                 const float* __restrict__ bias, int M) {
  long long idx = (long long)blockIdx.x * blockDim.x + threadIdx.x;
  if (idx >= (long long)M * HDIM) return;
  const int r = (int)(idx >> 7);
  const int f = (int)(idx & 127);
  out[(size_t)r * ldc + f] = bias[f];
}

// ------------------------------------------------------- WMMA f32 GEMM core
// out[M,128](ldc) = act( alpha * (A .* colscale) @ W^T + bias )
//   A: [M, Ka] (lda), W: [128, Ka] row-major. ACT: 0=none 1=relu 2=tanh.
// K-step 4 (V_WMMA_F32_16X16X4_F32). Each wave: 4 M-tiles x one N-tile,
// sharing the B fragment (4 WMMAs per B load). alpha applied at writeout.
// ISA 7.12.2 32-bit layouts: A(16x4): lanes0-15 M=lane {K0,K1}, lanes16-31
// M=lane-16 {K2,K3}. B(4x16): vgpr r, half h -> K=r+2h, N=lane%16.
// C/D(16x16): vgpr i, half h -> M=i+8h, N=lane%16.
template <int ACT, bool SCALE>
__global__ __launch_bounds__(128)
void wmma_gemm_kernel(const float* __restrict__ A, int lda,
                      const float* __restrict__ W,
                      const float* __restrict__ bias,
                      const float* __restrict__ colscale, float alpha,
                      float* __restrict__ out, int ldc, int M, int Ka) {
  const int lane = threadIdx.x;          // 0..31
  const int half = lane >> 4;            // 0/1
  const int lrow = lane & 15;
  const int tile_n = blockIdx.y;         // 0..7 (Nout = 128)
  const int mtiles = M >> 4;             // M % 16 == 0 for this model
  const int tm0 = (blockIdx.x * blockDim.y + threadIdx.y) * 4;
  if (tm0 >= mtiles) return;             // wave-uniform

  const float* brow = W + (size_t)(tile_n * 16 + lrow) * Ka;
  const float* arow = A + (size_t)(tm0 * 16 + lrow) * lda;
  const size_t as16 = (size_t)16 * lda;  // A stride for one M-tile

  const int col = tile_n * 16 + lrow;
  const float bv = bias ? bias[col] : 0.f;

  auto store_tile = [&](const v8f& c, int tm) {
#pragma unroll
    for (int i = 0; i < 8; ++i) {
      const int row = tm * 16 + half * 8 + i;
      float v = fmaf(c[i], alpha, bv);
      if (ACT == 1) v = fmaxf(v, 0.f);
      else if (ACT == 2) v = tanhf(v);
      out[(size_t)row * ldc + col] = v;
    }
  };

  if (tm0 + 4 <= mtiles) {
    // hot path: 4 M-tiles, B fragment reused 4x
    v8f c0 = {0.f, 0.f, 0.f, 0.f, 0.f, 0.f, 0.f, 0.f};
    v8f c1 = c0, c2 = c0, c3 = c0;
    for (int kk = 0; kk < Ka; kk += 4) {
      const int k0 = kk + half * 2;
      v2f b  = *(const v2f*)(brow + k0);
      v2f a0 = *(const v2f*)(arow + k0);
      v2f a1 = *(const v2f*)(arow + as16 + k0);
      v2f a2 = *(const v2f*)(arow + 2 * as16 + k0);
      v2f a3 = *(const v2f*)(arow + 3 * as16 + k0);
      if (SCALE) {
        const v2f s = *(const v2f*)(colscale + k0);
        a0.x *= s.x; a0.y *= s.y;
        a1.x *= s.x; a1.y *= s.y;
        a2.x *= s.x; a2.y *= s.y;
        a3.x *= s.x; a3.y *= s.y;
      }
      c0 = __builtin_amdgcn_wmma_f32_16x16x4_f32(false, a0, false, b, (short)0, c0, false, false);
      c1 = __builtin_amdgcn_wmma_f32_16x16x4_f32(false, a1, false, b, (short)0, c1, false, false);
      c2 = __builtin_amdgcn_wmma_f32_16x16x4_f32(false, a2, false, b, (short)0, c2, false, false);
      c3 = __builtin_amdgcn_wmma_f32_16x16x4_f32(false, a3, false, b, (short)0, c3, false, false);
    }
    store_tile(c0, tm0 + 0);
    store_tile(c1, tm0 + 1);
    store_tile(c2, tm0 + 2);
    store_tile(c3, tm0 + 3);
  } else {
    // tail: 1..3 M-tiles, one at a time (rare; wave-uniform)
    const int nt = mtiles - tm0;
    for (int t = 0; t < nt; ++t) {
      v8f c = {0.f, 0.f, 0.f, 0.f, 0.f, 0.f, 0.f, 0.f};
      const float* ar = arow + (size_t)t * as16;
      for (int kk = 0; kk < Ka; kk += 4) {
        const int k0 = kk + half * 2;
        v2f b = *(const v2f*)(brow + k0);
        v2f a = *(const v2f*)(ar + k0);
        if (SCALE) {
          const v2f s = *(const v2f*)(colscale + k0);
          a.x *= s.x; a.y *= s.y;
        }
        c = __builtin_amdgcn_wmma_f32_16x16x4_f32(false, a, false, b, (short)0, c, false, false);
      }
      store_tile(c, tm0 + t);
    }
  }
}

// ------------------------------------------------------------------- SpMM
// agg[dst] += feat[src] * dinv[src]   (segment_sum of m[src])
__global__ void spmm_kernel(const int* __restrict__ src, const int* __restrict__ dst,
                            const float* __restrict__ feat, const float* __restrict__ dinv,
                            float* __restrict__ agg, int E) {
  long long idx = (long long)blockIdx.x * blockDim.x + threadIdx.x;
  if (idx >= (long long)E * (HDIM / 4)) return;
  const int e = (int)(idx >> 5);        // HDIM/4 == 32 chunks per edge
  const int f = (int)(idx & 31) * 4;
  const int s = src[e], d = dst[e];
  const float dv = dinv[s];
  const float4 v = *(const float4*)(feat + (size_t)s * HDIM + f);
  float* o = agg + (size_t)d * HDIM + f;
  atomicAdd(o + 0, v.x * dv);
  atomicAdd(o + 1, v.y * dv);
  atomicAdd(o + 2, v.z * dv);
  atomicAdd(o + 3, v.w * dv);
}

// feat_out = feat_in - agg*dinv*diag ; hacc += theta*feat_out ; agg cleared
// (clearing here removes a dedicated zeroing pass before the next SpMM hop)
__global__ void lap_update_kernel(const float* __restrict__ fin, float* __restrict__ agg,
                                  const float* __restrict__ dinv, const float* __restrict__ diag,
                                  float theta, float* __restrict__ fout,
                                  float* __restrict__ hacc, int ldh, int n) {
  long long idx = (long long)blockIdx.x * blockDim.x + threadIdx.x;
  if (idx >= (long long)n * HDIM) return;
  const int node = (int)(idx >> 7);
  const int f = (int)(idx & 127);
  const float a = agg[idx];
  agg[idx] = 0.f;                        // re-establish zero invariant
  const float v = fin[idx] - a * dinv[node] * diag[f];
  fout[idx] = v;
  hacc[(size_t)node * ldh + f] += theta * v;
}

// ------------------------------------------------------------- attention
// wsum[comp] += sum_n dot(t[n,:], Wa2)   (wave-per-row, LDS block reduce)
__global__ void att_score_kernel(const float* __restrict__ t, const float* __restrict__ Wa2,
                                 float* __restrict__ wsum, int comp, int n) {
  __shared__ float bsum;
  if (threadIdx.x == 0) bsum = 0.f;
  __syncthreads();
  const int lane = threadIdx.x & 31;
  const int wave = (int)(((long long)blockIdx.x * blockDim.x + threadIdx.x) >> 5);
  float s = 0.f;
  if (wave < n) {
    const float4 v = *(const float4*)(t + (size_t)wave * HDIM + lane * 4);
    const float4 w = *(const float4*)(Wa2 + lane * 4);
    s = v.x * w.x + v.y * w.y + v.z * w.z + v.w * w.w;
  }
  for (int off = 16; off; off >>= 1) s += __shfl_xor(s, off);
  if (lane == 0 && wave < n) atomicAdd(&bsum, s);
  __syncthreads();
  if (threadIdx.x == 0) atomicAdd(&wsum[comp], bsum);
}

__global__ void beta_kernel(const float* __restrict__ wsum, float* __restrict__ beta, float invN) {
  if (threadIdx.x == 0 && blockIdx.x == 0) {
    const float m0 = wsum[0] * invN, m1 = wsum[1] * invN;
    const float mx = fmaxf(m0, m1);
    const float e0 = expf(m0 - mx), e1 = expf(m1 - mx);
    const float inv = 1.f / (e0 + e1);
    beta[0] = e0 * inv;
    beta[1] = e1 * inv;
  }
}

__global__ void emb_kernel(const float* __restrict__ h_o, const float* __restrict__ h_knn,
                           const float* __restrict__ beta, float* __restrict__ emb, int n) {
  long long idx = (long long)blockIdx.x * blockDim.x + threadIdx.x;
  if (idx < (long long)n * HDIM)
    emb[idx] = beta[0] * h_o[idx] + beta[1] * h_knn[idx];
}

// logits[n,0:2] = emb[n,:] @ W4^T + b4   (wave-per-row)
__global__ void logits_kernel(const float* __restrict__ emb, const float* __restrict__ W4,
                              const float* __restrict__ b4, float* __restrict__ logits, int n) {
  const int lane = threadIdx.x & 31;
  const int wave = (int)(((long long)blockIdx.x * blockDim.x + threadIdx.x) >> 5);
  if (wave >= n) return;
  const float4 e = *(const float4*)(emb + (size_t)wave * HDIM + lane * 4);
  const float4 w0 = *(const float4*)(W4 + lane * 4);
  const float4 w1 = *(const float4*)(W4 + HDIM + lane * 4);
  float s0 = e.x * w0.x + e.y * w0.y + e.z * w0.z + e.w * w0.w;
  float s1 = e.x * w1.x + e.y * w1.y + e.z * w1.z + e.w * w1.w;
  for (int off = 16; off; off >>= 1) {
    s0 += __shfl_xor(s0, off);
    s1 += __shfl_xor(s1, off);
  }
  if (lane == 0) {
    logits[(size_t)wave * 2 + 0] = s0 + b4[0];
    logits[(size_t)wave * 2 + 1] = s1 + b4[1];
  }
}

// ---------------------------------------------------------------- launcher
extern "C" void kernel_launch(void* const* d_in, const int* in_sizes, int n_in,
                              void* d_out, int out_size, void* d_ws, size_t ws_size,
                              hipStream_t stream) {
  const float* x      = (const float*)d_in[0];
  const int*   src_g  = (const int*)d_in[1];
  const int*   dst_g  = (const int*)d_in[2];
  const int*   src_k  = (const int*)d_in[3];
  const int*   dst_k  = (const int*)d_in[4];
  const float* W1     = (const float*)d_in[5];
  const float* b1     = (const float*)d_in[6];
  const float* W2     = (const float*)d_in[7];
  const float* b2     = (const float*)d_in[8];
  const float* diag_g = (const float*)d_in[9];
  const float* Wc_g   = (const float*)d_in[10];
  const float* bc_g   = (const float*)d_in[11];
  const float* diag_k = (const float*)d_in[12];
  const float* Wc_k   = (const float*)d_in[13];
  const float* bc_k   = (const float*)d_in[14];
  const float* W3     = (const float*)d_in[15];
  const float* b3     = (const float*)d_in[16];
  const float* Wk     = (const float*)d_in[17];
  const float* bk     = (const float*)d_in[18];
  const float* Wa1    = (const float*)d_in[19];
  const float* ba1    = (const float*)d_in[20];
  const float* Wa2    = (const float*)d_in[21];
  const float* W4     = (const float*)d_in[22];
  const float* b4     = (const float*)d_in[23];

  const int IN_F = 64;
  const int N  = in_sizes[0] / IN_F;   // 50000 (divisible by 16)
  const int EG = in_sizes[1];
  const int EK = in_sizes[3];

  // ---- workspace carve-up (float units; every offset multiple of N floats)
  float* ws     = (float*)d_ws;
  float* dinv_g = ws;                                  // N
  float* dinv_k = dinv_g + N;                          // N
  float* h      = dinv_k + N;                          // 128N  MLP output
  float* hcat   = h      + (size_t)N * HDIM;           // 384N  concat conv outs
  float* feat   = hcat   + (size_t)N * 3 * HDIM;       // 128N  Laplacian working feat
  float* agg    = feat   + (size_t)N * HDIM;           // 128N  SpMM acc / tanh scratch
  float* h_o    = agg    + (size_t)N * HDIM;           // 128N
  float* h_knn  = h_o    + (size_t)N * HDIM;           // 128N
  float* wsum   = h_knn  + (size_t)N * HDIM;           // 2
  float* beta   = wsum + 2;                            // 2

  float* logits = (float*)d_out;                       // [N,2]
  float* emb    = logits + (size_t)N * 2;              // [N,128]

  static const float TG[3][3] = {{3.f, -3.f, 0.75f}, {0.f, 3.f, -1.5f}, {0.f, 0.f, 0.75f}};
  static const float TK[3][3] = {{1.f, 1.f, 1.f}, {1.f, 1.f, 1.f}, {1.f, 1.f, 1.f}};

  auto zero = [&](float* p, long long cnt) {
    zero_f32_kernel<<<(unsigned)((cnt + 255) / 256), 256, 0, stream>>>(p, cnt);
  };
  auto gemm = [&](const float* A, int lda, const float* W, const float* bias,
                  const float* cs, float alpha, float* out, int ldc,
                  int M, int Ka, int act) {
    dim3 blk(32, 4);                                   // 4 waves / block
    const int mt = (M + 15) / 16;
    const int waves = (mt + 3) / 4;                    // 4 M-tiles per wave
    dim3 grd((unsigned)((waves + 3) / 4), HDIM / 16);
    if (cs)
      wmma_gemm_kernel<0, true><<<grd, blk, 0, stream>>>(A, lda, W, bias, cs, alpha, out, ldc, M, Ka);
    else if (act == 1)
      wmma_gemm_kernel<1, false><<<grd, blk, 0, stream>>>(A, lda, W, bias, cs, alpha, out, ldc, M, Ka);
    else if (act == 2)
      wmma_gemm_kernel<2, false><<<grd, blk, 0, stream>>>(A, lda, W, bias, cs, alpha, out, ldc, M, Ka);
    else
      wmma_gemm_kernel<0, false><<<grd, blk, 0, stream>>>(A, lda, W, bias, cs, alpha, out, ldc, M, Ka);
  };

  // ---- degrees: dinv = clip(deg,1)^-0.5 for both graphs
  zero(dinv_g, 2LL * N);
  degree_accum_kernel<<<(EG + 255) / 256, 256, 0, stream>>>(dst_g, dinv_g, EG);
  degree_accum_kernel<<<(EK + 255) / 256, 256, 0, stream>>>(dst_k, dinv_k, EK);
  finish_dinv_kernel<<<(2 * N + 255) / 256, 256, 0, stream>>>(dinv_g, 2 * N);

  // ---- input MLP
  gemm(x,    IN_F, W1, b1, nullptr, 1.f, feat, HDIM, N, IN_F, 1);
  gemm(feat, HDIM, W2, b2, nullptr, 1.f, h,    HDIM, N, HDIM, 1);

  // agg zero invariant: established once, maintained by lap_update_kernel
  zero(agg, (long long)N * HDIM);

  // ---- polyconv stacks for each graph
  auto run_graph = [&](const int* src, const int* dst, const float* dinv,
                       const float* diag, const float* Wc, const float* bc,
                       const float (*TH)[3], int E,
                       const float* Wf, const float* bf, float* hout) {
    for (int i = 0; i < 3; ++i) {
      // h_i = (theta0 * h * diag[i][0]) @ Wc[i]^T + bc[i]   -> hcat[:, i*H:(i+1)*H]
      if (TH[i][0] == 0.f) {
        long long nh = (long long)N * HDIM;
        fill_bias_kernel<<<(unsigned)((nh + 255) / 256), 256, 0, stream>>>(
            hcat + (size_t)i * HDIM, 3 * HDIM, bc + (size_t)i * HDIM, N);
      } else {
        gemm(h, HDIM, Wc + (size_t)i * HDIM * HDIM, bc + (size_t)i * HDIM,
             diag + (size_t)(i * 3 + 0) * HDIM, TH[i][0],
             hcat + (size_t)i * HDIM, 3 * HDIM, N, HDIM, 0);
      }
      const float* fsrc = h;
      for (int k = 1; k < 3; ++k) {
        long long tot = (long long)E * (HDIM / 4);
        spmm_kernel<<<(unsigned)((tot + 255) / 256), 256, 0, stream>>>(src, dst, fsrc, dinv, agg, E);
        long long nh = (long long)N * HDIM;
        lap_update_kernel<<<(unsigned)((nh + 255) / 256), 256, 0, stream>>>(
            fsrc, agg, dinv, diag + (size_t)(i * 3 + k) * HDIM, TH[i][k],
            feat, hcat + (size_t)i * HDIM, 3 * HDIM, N);
        fsrc = feat;
      }
    }
    // fusion: hout = relu(hcat @ Wf^T + bf)
    gemm(hcat, 3 * HDIM, Wf, bf, nullptr, 1.f, hout, HDIM, N, 3 * HDIM, 1);
  };

  run_graph(src_g, dst_g, dinv_g, diag_g, Wc_g, bc_g, TG, EG, W3, b3, h_o);
  run_graph(src_k, dst_k, dinv_k, diag_k, Wc_k, bc_k, TK, EK, Wk, bk, h_knn);

  // ---- component attention
  zero(wsum, 2);
  const unsigned rb = (unsigned)((N + 7) / 8);  // wave-per-row, 8 waves/block
  gemm(h_o, HDIM, Wa1, ba1, nullptr, 1.f, agg, HDIM, N, HDIM, 2 /*tanh*/);
  att_score_kernel<<<rb, 256, 0, stream>>>(agg, Wa2, wsum, 0, N);
  gemm(h_knn, HDIM, Wa1, ba1, nullptr, 1.f, agg, HDIM, N, HDIM, 2 /*tanh*/);
  att_score_kernel<<<rb, 256, 0, stream>>>(agg, Wa2, wsum, 1, N);
  beta_kernel<<<1, 32, 0, stream>>>(wsum, beta, 1.f / (float)N);

  long long nh = (long long)N * HDIM;
  emb_kernel<<<(unsigned)((nh + 255) / 256), 256, 0, stream>>>(h_o, h_knn, beta, emb, N);
  logits_kernel<<<rb, 256, 0, stream>>>(emb, W4, b4, logits, N);
}